// Mvdr_38843684225667
// MI455X (gfx1250) — compile-verified
//
#include <hip/hip_runtime.h>

typedef float v2f __attribute__((ext_vector_type(2)));
typedef float v8f __attribute__((ext_vector_type(8)));

#define B_SZ   8
#define C_CH   4
#define L_SMP  160000
#define HOP    160
#define NFFT   512
#define FB     257          // NFFT/2+1
#define NB2    514          // 2*FB
#define T_FR   1001         // L/HOP + 1
#define NSEQ   32           // B*C
#define HID    128
#define G4     512          // 4*HID
#define PAD    256          // NFFT/2

// ---------------- device helpers ----------------

__device__ __forceinline__ float hann512(int j) {
  // periodic Hann(320) zero-padded by 96 on each side to 512
  if (j < 96 || j >= 416) return 0.f;
  float ph = (float)(j - 96) * (6.28318530717958647692f / 320.f);
  return 0.5f * (1.f - __cosf(ph));
}
__device__ __forceinline__ float sigm(float x) { return 1.f / (1.f + __expf(-x)); }
__device__ __forceinline__ float tanh_fast(float x) {
  x = fminf(15.f, fmaxf(-15.f, x));
  float e = __expf(2.f * x);
  return (e - 1.f) / (e + 1.f);
}
__device__ __forceinline__ v8f wmma4(v2f a, v2f b, v8f c) {
  // D = A(16x4,f32) x B(4x16,f32) + C(16x16,f32)
  return __builtin_amdgcn_wmma_f32_16x16x4_f32(false, a, false, b, (short)0, c,
                                               false, false);
}

// ---------------- setup kernels ----------------

// Forward rDFT basis Bf (512 x 514): [cos | -sin]; inverse basis IB (514 x 512).
__global__ void k_basis(float* __restrict__ Bf, float* __restrict__ IB) {
  int idx = blockIdx.x * blockDim.x + threadIdx.x;
  const int tot1 = NFFT * NB2;
  if (idx < tot1) {
    int j = idx / NB2, k = idx % NB2;
    int kk = (k < FB) ? k : (k - FB);
    int m = (j * kk) & (NFFT - 1);                 // exact phase reduction
    float ph = (float)m * (6.28318530717958647692f / (float)NFFT);
    Bf[idx] = (k < FB) ? __cosf(ph) : -__sinf(ph);
  } else if (idx < 2 * tot1) {
    int r = idx - tot1;
    int k = r / NFFT, j = r % NFFT;                // IB row k, col j
    int kk = (k < FB) ? k : (k - FB);
    float cs = ((kk == 0) || (kk == NFFT / 2)) ? 1.f : 2.f;
    int m = (j * kk) & (NFFT - 1);
    float ph = (float)m * (6.28318530717958647692f / (float)NFFT);
    float v = (k < FB) ? __cosf(ph) : -__sinf(ph);
    IB[r] = v * cs * (1.f / (float)NFFT);
  }
}

__global__ void k_bias(const float* __restrict__ a, const float* __restrict__ b,
                       float* __restrict__ o, int n) {
  int i = blockIdx.x * blockDim.x + threadIdx.x;
  if (i < n) o[i] = a[i] + b[i];
}

// Windowed frames with reflect padding: frames[(n*T+t)*512 + j]
__global__ void k_frames(const float* __restrict__ x, float* __restrict__ frames) {
  long idx = (long)blockIdx.x * blockDim.x + threadIdx.x;
  const long tot = (long)NSEQ * T_FR * NFFT;
  if (idx >= tot) return;
  int j = (int)(idx % NFFT);
  long mt = idx / NFFT;
  int t = (int)(mt % T_FR);
  int n = (int)(mt / T_FR);
  int b = n >> 2, c = n & 3;
  int pos = t * HOP + j - PAD;
  if (pos < 0) pos = -pos;                         // reflect
  if (pos >= L_SMP) pos = 2 * L_SMP - 2 - pos;
  float v = x[((size_t)b * L_SMP + pos) * C_CH + c];   // x is (B,L,C)
  frames[idx] = v * hann512(j);
}

// ---------------- generic fp32 WMMA GEMM (register-blocked) ----------------
// C(MxN) = A(MxK) x B + bias.  TB=0: B is (KxN); TB=1: B is (NxK) (weights).
// Block = 128 threads (4 waves); block tile = 16(M) x 256(N); each wave owns a
// 16x64 strip (4 accumulators) so one A-fragment load feeds 4 WMMAs.
template <int TB>
__global__ void __launch_bounds__(128)
k_gemm(const float* __restrict__ A, const float* __restrict__ Bm,
       const float* __restrict__ bias, float* __restrict__ C,
       int M, int N, int K) {
  const int lane = threadIdx.x & 31;
  const int wv = threadIdx.x >> 5;
  const int mrow = lane & 15;
  const int half = lane >> 4;                      // A/B fragment K half
  const int m0 = blockIdx.x * 16;
  const int nbase = blockIdx.y * 256 + wv * 64;    // wave strip: 4 tiles of 16
  int m = m0 + mrow;
  if (m >= M) m = M - 1;                           // clamp loads, guard stores

  int nc[4], ncl[4];
  bool nval[4];
#pragma unroll
  for (int s = 0; s < 4; ++s) {
    nc[s] = nbase + s * 16 + mrow;
    nval[s] = nc[s] < N;
    ncl[s] = nval[s] ? nc[s] : 0;
  }

  v8f acc[4];
#pragma unroll
  for (int s = 0; s < 4; ++s)
#pragma unroll
    for (int r = 0; r < 8; ++r) acc[s][r] = 0.f;

  const size_t arow = (size_t)m * K;
  const int K4 = K & ~3;
  for (int k0 = 0; k0 < K4; k0 += 4) {
    const int koff = k0 + 2 * half;
    v2f a = *(const v2f*)(A + arow + koff);        // loaded once, used 4x
#pragma unroll
    for (int s = 0; s < 4; ++s) {
      v2f b;
      if (TB) {
        b = *(const v2f*)(Bm + (size_t)ncl[s] * K + koff);
      } else {
        b[0] = Bm[(size_t)koff * N + ncl[s]];
        b[1] = Bm[(size_t)(koff + 1) * N + ncl[s]];
      }
      acc[s] = wmma4(a, b, acc[s]);
    }
  }
  if (K4 < K) {                                    // K % 4 == 2 (K = 514)
    const int koff = K4 + 2 * half;
    v2f a;
    a[0] = (koff < K) ? A[arow + koff] : 0.f;
    a[1] = (koff + 1 < K) ? A[arow + koff + 1] : 0.f;
#pragma unroll
    for (int s = 0; s < 4; ++s) {
      v2f b;
      if (TB) {
        b[0] = (koff < K) ? Bm[(size_t)ncl[s] * K + koff] : 0.f;
        b[1] = (koff + 1 < K) ? Bm[(size_t)ncl[s] * K + koff + 1] : 0.f;
      } else {
        b[0] = (koff < K) ? Bm[(size_t)koff * N + ncl[s]] : 0.f;
        b[1] = (koff + 1 < K) ? Bm[(size_t)(koff + 1) * N + ncl[s]] : 0.f;
      }
      acc[s] = wmma4(a, b, acc[s]);
    }
  }

#pragma unroll
  for (int s = 0; s < 4; ++s) {
    const float bs = (bias != nullptr && nval[s]) ? bias[ncl[s]] : 0.f;
#pragma unroll
    for (int r = 0; r < 8; ++r) {
      const int row = m0 + r + 8 * half;           // C/D layout: vgpr r -> M=r / M=8+r
      if (row < M && nval[s]) C[(size_t)row * N + nc[s]] = acc[s][r] + bs;
    }
  }
}

// ---------------- persistent-LDS LSTM recurrence ----------------
// One workgroup (8 wave32). W_hh (512x128, 256KB) + h (32x128, 16KB) live in the
// 320KB WGP LDS. Wave w owns hidden columns [16w,16w+16) of all four gates, so
// the i/f/g/o fusion and the c/h update are wave-private. Two barriers/step.
// The per-step xw gate biases are loaded BEFORE the recurrent GEMM so their
// global latency overlaps the WMMA chain (software pipelining on the serial path).
__global__ void __launch_bounds__(256, 1)
k_lstm(const float* __restrict__ xw, const float* __restrict__ Whh,
       float* __restrict__ hout) {
  extern __shared__ float smem[];
  float* Whh_s = smem;                             // 512*128
  float* h_s = smem + G4 * HID;                    // 32*128
  const int tid = threadIdx.x;
  for (int i = tid; i < G4 * HID; i += 256) Whh_s[i] = Whh[i];
  for (int i = tid; i < NSEQ * HID; i += 256) h_s[i] = 0.f;

  const int lane = tid & 31, wv = tid >> 5;
  const int mrow = lane & 15, half = lane >> 4;
  const int hcol = wv * 16 + mrow;

  v8f cst[2];
#pragma unroll
  for (int mt = 0; mt < 2; ++mt)
#pragma unroll
    for (int r = 0; r < 8; ++r) cst[mt][r] = 0.f;

  __syncthreads();
  for (int t = 0; t < T_FR; ++t) {
    // -- issue the xw loads for this step first (independent of the GEMM) --
    float xv[2][4][8];
#pragma unroll
    for (int mt = 0; mt < 2; ++mt)
#pragma unroll
      for (int r = 0; r < 8; ++r) {
        const int seq = mt * 16 + r + 8 * half;
        const float* xrow = xw + (size_t)seq * T_FR * G4 + (size_t)t * G4;
#pragma unroll
        for (int gt = 0; gt < 4; ++gt)
          xv[mt][gt][r] = xrow[gt * HID + hcol];
      }

    v8f acc[8];                                    // [mt*4 + gate]
#pragma unroll
    for (int i = 0; i < 8; ++i)
#pragma unroll
      for (int r = 0; r < 8; ++r) acc[i][r] = 0.f;

    for (int k0 = 0; k0 < HID; k0 += 4) {          // h @ W_hh^T, K = 128
      const int koff = k0 + 2 * half;
      v2f a0 = *(const v2f*)&h_s[mrow * HID + koff];
      v2f a1 = *(const v2f*)&h_s[(16 + mrow) * HID + koff];
#pragma unroll
      for (int gt = 0; gt < 4; ++gt) {
        v2f b = *(const v2f*)&Whh_s[(gt * HID + hcol) * HID + koff];
        acc[gt] = wmma4(a0, b, acc[gt]);
        acc[4 + gt] = wmma4(a1, b, acc[4 + gt]);
      }
    }

    v8f hnew[2];
#pragma unroll
    for (int mt = 0; mt < 2; ++mt) {
#pragma unroll
      for (int r = 0; r < 8; ++r) {
        const float gi = acc[mt * 4 + 0][r] + xv[mt][0][r];
        const float gf = acc[mt * 4 + 1][r] + xv[mt][1][r];
        const float gg = acc[mt * 4 + 2][r] + xv[mt][2][r];
        const float go = acc[mt * 4 + 3][r] + xv[mt][3][r];
        const float cv = sigm(gf) * cst[mt][r] + sigm(gi) * tanh_fast(gg);
        cst[mt][r] = cv;
        hnew[mt][r] = sigm(go) * tanh_fast(cv);
      }
    }
    __syncthreads();                               // all h_s reads done
#pragma unroll
    for (int mt = 0; mt < 2; ++mt)
#pragma unroll
      for (int r = 0; r < 8; ++r) {
        const int seq = mt * 16 + r + 8 * half;
        const float hv = hnew[mt][r];
        h_s[seq * HID + hcol] = hv;
        hout[(size_t)seq * T_FR * HID + (size_t)t * HID + hcol] = hv;
      }
    __syncthreads();                               // writes visible next step
  }
}

// ---------------- PSD accumulation ----------------
// One block per (b,f): accumulate 4x4 complex psd_s / psd_n over t.
__global__ void k_psd(const float* __restrict__ feats, const float* __restrict__ mk0,
                      const float* __restrict__ mk1, float* __restrict__ psd_s,
                      float* __restrict__ psd_n) {
  const int bf = blockIdx.x;
  const int b = bf / FB, f = bf % FB;
  const int tid = threadIdx.x;                     // 0..127
  float sre[16], smi[16], nre[16], nim[16];
#pragma unroll
  for (int i = 0; i < 16; ++i) { sre[i] = 0; smi[i] = 0; nre[i] = 0; nim[i] = 0; }

  for (int t = tid; t < T_FR; t += 128) {
    float s_r[4], s_i[4], n_r[4], n_i[4];
#pragma unroll
    for (int c = 0; c < 4; ++c) {
      const size_t base = ((size_t)(b * 4 + c) * T_FR + t) * NB2;
      const float r = feats[base + f], ii = feats[base + FB + f];
      float rm = mk0[base + f], im = mk0[base + FB + f];
      s_r[c] = rm * r - im * ii; s_i[c] = rm * ii + im * r;
      rm = mk1[base + f]; im = mk1[base + FB + f];
      n_r[c] = rm * r - im * ii; n_i[c] = rm * ii + im * r;
    }
#pragma unroll
    for (int c = 0; c < 4; ++c)
#pragma unroll
      for (int d = 0; d < 4; ++d) {                // s_c * conj(s_d)
        const int i = c * 4 + d;
        sre[i] += s_r[c] * s_r[d] + s_i[c] * s_i[d];
        smi[i] += s_i[c] * s_r[d] - s_r[c] * s_i[d];
        nre[i] += n_r[c] * n_r[d] + n_i[c] * n_i[d];
        nim[i] += n_i[c] * n_r[d] - n_r[c] * n_i[d];
      }
  }
#pragma unroll
  for (int i = 0; i < 16; ++i)
    for (int off = 16; off; off >>= 1) {
      sre[i] += __shfl_xor(sre[i], off, 32);
      smi[i] += __shfl_xor(smi[i], off, 32);
      nre[i] += __shfl_xor(nre[i], off, 32);
      nim[i] += __shfl_xor(nim[i], off, 32);
    }
  __shared__ float red[4][64];
  const int lane = tid & 31, wv = tid >> 5;
  if (lane == 0) {
#pragma unroll
    for (int i = 0; i < 16; ++i) {
      red[wv][i] = sre[i]; red[wv][16 + i] = smi[i];
      red[wv][32 + i] = nre[i]; red[wv][48 + i] = nim[i];
    }
  }
  __syncthreads();
  if (tid < 64) {
    float v = red[0][tid] + red[1][tid] + red[2][tid] + red[3][tid];
    const int kind = tid >> 4, i = tid & 15;
    const int c = i >> 2, d = i & 3;
    const size_t base = ((size_t)bf * 16 + i) * 2;
    if (kind == 0) psd_s[base] = v;
    else if (kind == 1) psd_s[base + 1] = v;
    else if (kind == 2) psd_n[base] = v + ((c == d) ? 1e-6f : 0.f);
    else psd_n[base + 1] = v;
  }
}

// ---------------- MVDR 4x4 complex solve per (b,f) ----------------
__global__ void k_mvdr(const float* __restrict__ psdS, const float* __restrict__ psdN,
                       float* __restrict__ u) {
  const int bf = blockIdx.x * blockDim.x + threadIdx.x;
  if (bf >= B_SZ * FB) return;
  float Nr[4][4], Ni[4][4], Sr[4][4], Si[4][4];
  const size_t base = (size_t)bf * 32;
#pragma unroll
  for (int i = 0; i < 16; ++i) {
    Sr[i >> 2][i & 3] = psdS[base + 2 * i];
    Si[i >> 2][i & 3] = psdS[base + 2 * i + 1];
    Nr[i >> 2][i & 3] = psdN[base + 2 * i];
    Ni[i >> 2][i & 3] = psdN[base + 2 * i + 1];
  }
  // [N | S] -> [I | X], X = N^{-1} S (diagonally-loaded Hermitian, no pivoting)
#pragma unroll
  for (int p = 0; p < 4; ++p) {
    const float pr = Nr[p][p], pi = Ni[p][p];
    const float dn = pr * pr + pi * pi + 1e-30f;
    const float ir = pr / dn, ii = -pi / dn;
#pragma unroll
    for (int j = 0; j < 4; ++j) {
      float xr = Nr[p][j], xi = Ni[p][j];
      Nr[p][j] = xr * ir - xi * ii; Ni[p][j] = xr * ii + xi * ir;
      xr = Sr[p][j]; xi = Si[p][j];
      Sr[p][j] = xr * ir - xi * ii; Si[p][j] = xr * ii + xi * ir;
    }
#pragma unroll
    for (int r = 0; r < 4; ++r)
      if (r != p) {
        const float fr = Nr[r][p], fi = Ni[r][p];
#pragma unroll
        for (int j = 0; j < 4; ++j) {
          Nr[r][j] -= fr * Nr[p][j] - fi * Ni[p][j];
          Ni[r][j] -= fr * Ni[p][j] + fi * Nr[p][j];
          Sr[r][j] -= fr * Sr[p][j] - fi * Si[p][j];
          Si[r][j] -= fr * Si[p][j] + fi * Sr[p][j];
        }
      }
  }
  const float tr = Sr[0][0] + Sr[1][1] + Sr[2][2] + Sr[3][3] + 1e-15f;
  const float ti = Si[0][0] + Si[1][1] + Si[2][2] + Si[3][3];
  const float dd = tr * tr + ti * ti;
#pragma unroll
  for (int c = 0; c < 4; ++c) {                    // u_c = X[c][0] / tr (REF_CH=0)
    const float xr = Sr[c][0], xi = Si[c][0];
    u[((size_t)bf * 4 + c) * 2] = (xr * tr + xi * ti) / dd;
    u[((size_t)bf * 4 + c) * 2 + 1] = (xi * tr - xr * ti) / dd;
  }
}

// ---------------- beamform: enh[b,t,f] = sum_c conj(u) * spec ----------------
__global__ void k_beamform(const float* __restrict__ feats, const float* __restrict__ u,
                           float* __restrict__ enhE) {
  long idx = (long)blockIdx.x * blockDim.x + threadIdx.x;
  const long tot = (long)B_SZ * T_FR * FB;
  if (idx >= tot) return;
  int f = (int)(idx % FB);
  long bt = idx / FB;
  int t = (int)(bt % T_FR);
  int b = (int)(bt / T_FR);
  float er = 0.f, ei = 0.f;
#pragma unroll
  for (int c = 0; c < 4; ++c) {
    const size_t base = ((size_t)(b * 4 + c) * T_FR + t) * NB2;
    const float r = feats[base + f], im = feats[base + FB + f];
    const size_t ub = ((size_t)(b * FB + f) * 4 + c) * 2;
    const float ur = u[ub], ui = u[ub + 1];
    er += ur * r + ui * im;                        // conj(u) * spec
    ei += ur * im - ui * r;
  }
  const size_t ob = ((size_t)b * T_FR + t) * NB2;
  enhE[ob + f] = er;
  enhE[ob + FB + f] = ei;
}

// ---------------- overlap-add + window^2 normalization (gather, no atomics) ----
__global__ void k_ola(const float* __restrict__ frT, float* __restrict__ out) {
  long idx = (long)blockIdx.x * blockDim.x + threadIdx.x;
  const long tot = (long)B_SZ * L_SMP;
  if (idx >= tot) return;
  int p = (int)(idx % L_SMP);
  int b = (int)(idx / L_SMP);
  const int q = p + PAD;
  int tmin = (q > (NFFT - 1)) ? ((q - (NFFT - 1) + HOP - 1) / HOP) : 0;
  int tmax = q / HOP;
  if (tmax > T_FR - 1) tmax = T_FR - 1;
  float num = 0.f, den = 0.f;
  for (int t = tmin; t <= tmax; ++t) {
    const int j = q - t * HOP;
    const float w = hann512(j);
    num += frT[((size_t)b * T_FR + t) * NFFT + j] * w;
    den += w * w;
  }
  out[idx] = num / ((den > 1e-11f) ? den : 1.f);
}

// ---------------- host launch ----------------
extern "C" void kernel_launch(void* const* d_in, const int* in_sizes, int n_in,
                              void* d_out, int out_size, void* d_ws, size_t ws_size,
                              hipStream_t stream) {
  const float* x = (const float*)d_in[0];
  const float* wih0 = (const float*)d_in[1];
  const float* whh0 = (const float*)d_in[2];
  const float* bih0 = (const float*)d_in[3];
  const float* bhh0 = (const float*)d_in[4];
  const float* wih1 = (const float*)d_in[5];
  const float* whh1 = (const float*)d_in[6];
  const float* bih1 = (const float*)d_in[7];
  const float* bhh1 = (const float*)d_in[8];
  const float* l0w = (const float*)d_in[9];
  const float* l0b = (const float*)d_in[10];
  const float* l1w = (const float*)d_in[11];
  const float* l1b = (const float*)d_in[12];
  float* out = (float*)d_out;
  float* ws = (float*)d_ws;

  const size_t M32 = (size_t)NSEQ * T_FR;          // 32032 = 2002 * 16
  const size_t M8 = (size_t)B_SZ * T_FR;           // 8008

  size_t o = 0;
  float* Bf = ws + o;     o += (size_t)NFFT * NB2;
  float* IB = ws + o;     o += (size_t)NB2 * NFFT;
  float* bias0 = ws + o;  o += G4;
  float* bias1 = ws + o;  o += G4;
  float* frames = ws + o; o += M32 * NFFT;         // reused as xw
  float* feats = ws + o;  o += M32 * NB2;          // spec: [Re(257) | Im(257)]
  float* h0 = ws + o;     o += M32 * HID;
  float* h1 = ws + o;     o += M32 * HID;
  float* mask0 = ws + o;  o += M32 * NB2;
  float* mask1 = ws + o;  o += M32 * NB2;
  float* psdS = ws + o;   o += (size_t)B_SZ * FB * 32;
  float* psdN = ws + o;   o += (size_t)B_SZ * FB * 32;
  float* ubuf = ws + o;   o += (size_t)B_SZ * FB * 8;
  float* enhE = ws + o;   o += M8 * NB2;
  float* frT = ws + o;    o += M8 * NFFT;
  float* xw = frames;

  // 0) DFT bases + fused biases
  {
    int tot = 2 * NFFT * NB2;
    k_basis<<<(tot + 255) / 256, 256, 0, stream>>>(Bf, IB);
  }
  k_bias<<<(G4 + 255) / 256, 256, 0, stream>>>(bih0, bhh0, bias0, G4);
  k_bias<<<(G4 + 255) / 256, 256, 0, stream>>>(bih1, bhh1, bias1, G4);

  // 1) windowed frames (reflect pad)
  {
    long tot = (long)M32 * NFFT;
    k_frames<<<(unsigned)((tot + 255) / 256), 256, 0, stream>>>(x, frames);
  }
  // 2) STFT GEMM: feats(32032x514) = frames(32032x512) x Bf(512x514)
  {
    dim3 g((unsigned)((M32 + 15) / 16), (NB2 + 255) / 256);
    k_gemm<0><<<g, 128, 0, stream>>>(frames, Bf, nullptr, feats, (int)M32, NB2, NFFT);
  }
  // 3) xw0 = feats x W_ih0^T + (b_ih0 + b_hh0)
  {
    dim3 g((unsigned)((M32 + 15) / 16), (G4 + 255) / 256);
    k_gemm<1><<<g, 128, 0, stream>>>(feats, wih0, bias0, xw, (int)M32, G4, NB2);
  }
  // 4) LSTM layer 0 (persistent, 272KB LDS of the 320KB WGP pool)
  const size_t lstm_shmem = (size_t)(G4 * HID + NSEQ * HID) * sizeof(float);
  k_lstm<<<1, 256, lstm_shmem, stream>>>(xw, whh0, h0);
  // 5) xw1 = h0 x W_ih1^T + (b_ih1 + b_hh1)
  {
    dim3 g((unsigned)((M32 + 15) / 16), (G4 + 255) / 256);
    k_gemm<1><<<g, 128, 0, stream>>>(h0, wih1, bias1, xw, (int)M32, G4, HID);
  }
  // 6) LSTM layer 1
  k_lstm<<<1, 256, lstm_shmem, stream>>>(xw, whh1, h1);
  // 7) mask heads
  {
    dim3 g((unsigned)((M32 + 15) / 16), (NB2 + 255) / 256);
    k_gemm<1><<<g, 128, 0, stream>>>(h1, l0w, l0b, mask0, (int)M32, NB2, HID);
    k_gemm<1><<<g, 128, 0, stream>>>(h1, l1w, l1b, mask1, (int)M32, NB2, HID);
  }
  // 8) per-bin spatial covariances
  k_psd<<<B_SZ * FB, 128, 0, stream>>>(feats, mask0, mask1, psdS, psdN);
  // 9) MVDR weights
  k_mvdr<<<(B_SZ * FB + 63) / 64, 64, 0, stream>>>(psdS, psdN, ubuf);
  // 10) beamform
  {
    long tot = (long)B_SZ * T_FR * FB;
    k_beamform<<<(unsigned)((tot + 255) / 256), 256, 0, stream>>>(feats, ubuf, enhE);
  }
  // 11) iSTFT GEMM: frT(8008x512) = enhE(8008x514) x IB(514x512)
  {
    dim3 g((unsigned)((M8 + 15) / 16), (NFFT + 255) / 256);
    k_gemm<0><<<g, 128, 0, stream>>>(enhE, IB, nullptr, frT, (int)M8, NFFT, NB2);
  }
  // 12) overlap-add + normalization + crop
  {
    long tot = (long)B_SZ * L_SMP;
    k_ola<<<(unsigned)((tot + 255) / 256), 256, 0, stream>>>(frT, out);
  }
  (void)in_sizes; (void)n_in; (void)out_size; (void)ws_size;
}